// GAT1_3315714753149
// MI455X (gfx1250) — compile-verified
//
#include <hip/hip_runtime.h>

// GAT layer pipeline for gfx1250 (MI455X), compile-only target.
// H=4 heads, C=128 channels, D_in=8, N_GRAPHS=64.

typedef float v2f __attribute__((ext_vector_type(2)));
typedef float v8f __attribute__((ext_vector_type(8)));

#define HC 512   // H*C
#define CC 128   // C
#define DIN 8

// ---- monotone float <-> uint encoding for atomic max on floats ----
__device__ __forceinline__ unsigned f2u_mono(float f) {
  unsigned u = __float_as_uint(f);
  return (u & 0x80000000u) ? ~u : (u | 0x80000000u);
}
__device__ __forceinline__ float u2f_mono(unsigned u) {
  u = (u & 0x80000000u) ? (u ^ 0x80000000u) : ~u;
  return __uint_as_float(u);
}

// ---------------------------------------------------------------------------
// Generic skinny GEMM  Dst[N x ncols] = A[N x 8] * B[8 x ncols]
// via two chained V_WMMA_F32_16X16X4_F32 per 16x16 tile (one wave per tile).
// B addressing is strided so both W_gat (row-major [8 x 512]) and
// lin_w^T (B[k][c] = lin_w[c*8 + k]) reuse this kernel.
// A layout (16x4 f32): lane l -> m = l&15, ks = l>>4; holds K = 2*ks, 2*ks+1.
// B layout (4x16 f32): lane l -> n = l&15, ks = l>>4; holds K = 2*ks, 2*ks+1.
// C/D (16x16 f32, 8 VGPRs): lane l, reg r -> row m = r + 8*(l>>4), col = l&15.
// ---------------------------------------------------------------------------
__global__ __launch_bounds__(32) void k_gemm8(const float* __restrict__ A,
                                              const float* __restrict__ B,
                                              float* __restrict__ Dst, int N,
                                              int out_stride, int b_row_stride,
                                              int b_col_stride) {
  const int lane = threadIdx.x;
  const int m    = lane & 15;
  const int ks   = lane >> 4;                 // 0 or 1
  const int row  = blockIdx.x * 16 + m;
  const int col  = blockIdx.y * 16 + m;       // (lane&15) doubles as B/D column
  const bool full_tile = (blockIdx.x * 16 + 15) < N;  // block-uniform

  v2f a0 = {0.f, 0.f}, a1 = {0.f, 0.f};
  if (row < N) {
    const float* ar = A + (size_t)row * DIN;
    a0.x = ar[2 * ks + 0];
    a0.y = ar[2 * ks + 1];
    a1.x = ar[4 + 2 * ks + 0];
    a1.y = ar[4 + 2 * ks + 1];
  }
  v2f b0, b1;
  const int cb = col * b_col_stride;
  b0.x = B[(2 * ks + 0) * b_row_stride + cb];
  b0.y = B[(2 * ks + 1) * b_row_stride + cb];
  b1.x = B[(4 + 2 * ks + 0) * b_row_stride + cb];
  b1.y = B[(4 + 2 * ks + 1) * b_row_stride + cb];

  v8f c = {};
  c = __builtin_amdgcn_wmma_f32_16x16x4_f32(false, a0, false, b0, (short)0, c,
                                            false, false);
  c = __builtin_amdgcn_wmma_f32_16x16x4_f32(false, a1, false, b1, (short)0, c,
                                            false, false);

  const int base_row = blockIdx.x * 16 + 8 * ks;
  float* o = Dst + (size_t)base_row * out_stride + col;
  if (full_tile) {
#pragma unroll
    for (int r = 0; r < 8; ++r) o[(size_t)r * out_stride] = c[r];
  } else {
#pragma unroll
    for (int r = 0; r < 8; ++r)
      if (base_row + r < N) o[(size_t)r * out_stride] = c[r];
  }
}

// ---------------------------------------------------------------------------
// Kernel 2: per-(node,head) attention logits a_src, a_dst (dots of length 128)
// ---------------------------------------------------------------------------
__global__ void k_att(const float* __restrict__ xh,
                      const float* __restrict__ att_s,
                      const float* __restrict__ att_d,
                      float* __restrict__ a_s, float* __restrict__ a_d, int NH) {
  int i = blockIdx.x * blockDim.x + threadIdx.x;
  if (i >= NH) return;
  int h = i & 3;
  const float* row = xh + (size_t)i * CC;   // i = n*4 + h indexes [N,H,C]
  const float* as = att_s + h * CC;
  const float* ad = att_d + h * CC;
  float ss = 0.f, sd = 0.f;
#pragma unroll 8
  for (int c = 0; c < CC; c += 4) {
    float4 v = *(const float4*)(row + c);
    float4 s = *(const float4*)(as + c);
    float4 d = *(const float4*)(ad + c);
    ss += v.x * s.x + v.y * s.y + v.z * s.z + v.w * s.w;
    sd += v.x * d.x + v.y * d.y + v.z * d.z + v.w * d.w;
  }
  a_s[i] = ss;
  a_d[i] = sd;
}

// ---------------------------------------------------------------------------
// Kernel 3: e = leaky_relu(a_src[src]+a_dst[dst]); segment-max over dst
// ---------------------------------------------------------------------------
__global__ void k_edge_max(const long long* __restrict__ ei,
                           const float* __restrict__ a_s,
                           const float* __restrict__ a_d,
                           float* __restrict__ e_buf,
                           unsigned* __restrict__ emax_u, long long E) {
  long long e = (long long)blockIdx.x * blockDim.x + threadIdx.x;
  if (e >= E) return;
  int src = (int)ei[e];
  int dst = (int)ei[E + e];
#pragma unroll
  for (int h = 0; h < 4; ++h) {
    float v = a_s[src * 4 + h] + a_d[dst * 4 + h];
    v = v > 0.f ? v : 0.2f * v;
    e_buf[e * 4 + h] = v;
    atomicMax(&emax_u[dst * 4 + h], f2u_mono(v));
  }
}

// ---------------------------------------------------------------------------
// Kernel 4: alpha_un = exp(e - emax[dst]); denom[dst] += alpha_un
// ---------------------------------------------------------------------------
__global__ void k_edge_exp(const long long* __restrict__ ei,
                           const unsigned* __restrict__ emax_u,
                           float* __restrict__ e_buf,
                           float* __restrict__ denom, long long E) {
  long long e = (long long)blockIdx.x * blockDim.x + threadIdx.x;
  if (e >= E) return;
  int dst = (int)ei[E + e];
#pragma unroll
  for (int h = 0; h < 4; ++h) {
    float m = u2f_mono(emax_u[dst * 4 + h]);
    float a = __expf(e_buf[e * 4 + h] - m);
    e_buf[e * 4 + h] = a;
    atomicAdd(&denom[dst * 4 + h], a);
  }
}

// ---------------------------------------------------------------------------
// Kernel 5: one wave per edge. Normalize alpha (write to output) and scatter
// the head-mean message: out_mean[dst] += 0.25 * sum_h alpha_h * xh[src,h,:].
// Lane l owns channels 4l..4l+3 (float4). x_h gathers are L2-resident (41 MB
// fits the 192 MB global L2), so this phase runs at L2 rather than HBM speed.
// ---------------------------------------------------------------------------
__global__ __launch_bounds__(256) void k_edge_aggr(
    const long long* __restrict__ ei, const float* __restrict__ xh,
    const float* __restrict__ alpha_un, const float* __restrict__ denom,
    float* __restrict__ alpha_out, float* __restrict__ out_mean, long long E) {
  const int lane = threadIdx.x & 31;
  long long e = (long long)blockIdx.x * (blockDim.x >> 5) + (threadIdx.x >> 5);
  if (e >= E) return;
  const int src = (int)ei[e];
  const int dst = (int)ei[E + e];

  float al[4];
#pragma unroll
  for (int h = 0; h < 4; ++h)
    al[h] = alpha_un[e * 4 + h] / (denom[dst * 4 + h] + 1e-16f);
  if (lane < 4) alpha_out[e * 4 + lane] = al[lane];

  const int c0 = lane * 4;
  float4 acc = {0.f, 0.f, 0.f, 0.f};
#pragma unroll
  for (int h = 0; h < 4; ++h) {
    float4 v = *(const float4*)(xh + (size_t)src * HC + h * CC + c0);
    acc.x += al[h] * v.x;
    acc.y += al[h] * v.y;
    acc.z += al[h] * v.z;
    acc.w += al[h] * v.w;
  }
  float* o = out_mean + (size_t)dst * CC + c0;
  atomicAdd(o + 0, 0.25f * acc.x);
  atomicAdd(o + 1, 0.25f * acc.y);
  atomicAdd(o + 2, 0.25f * acc.z);
  atomicAdd(o + 3, 0.25f * acc.w);
}

// ---------------------------------------------------------------------------
// Kernel 6: h = relu(out_mean + gat_bias + lin_buf + lin_b); BN partial sums.
// One channel per lane (128-lane block), 32 nodes per block -> register
// accumulation, 2 atomics per (block, channel).
// ---------------------------------------------------------------------------
__global__ __launch_bounds__(128) void k_node_fin(
    const float* __restrict__ out_mean, const float* __restrict__ lin_buf,
    const float* __restrict__ gat_bias, const float* __restrict__ lin_b,
    float* __restrict__ h_buf, float* __restrict__ bn_sum,
    float* __restrict__ bn_sumsq, int N, int rows_per_blk) {
  const int c = threadIdx.x;  // 0..127
  const int n0 = blockIdx.x * rows_per_blk;
  const float gb = gat_bias[c] + lin_b[c];
  float s = 0.f, s2 = 0.f;
  for (int i = 0; i < rows_per_blk; ++i) {
    int n = n0 + i;
    if (n >= N) break;
    size_t idx = (size_t)n * CC + c;
    float v = out_mean[idx] + gb + lin_buf[idx];
    v = v > 0.f ? v : 0.f;
    h_buf[idx] = v;
    s += v;
    s2 += v * v;
  }
  atomicAdd(&bn_sum[c], s);
  atomicAdd(&bn_sumsq[c], s2);
}

// ---------------------------------------------------------------------------
// Kernel 7: BN normalize + per-graph segment max (encoded uint atomicMax)
// ---------------------------------------------------------------------------
__global__ void k_bn_pool(const float* __restrict__ h_buf,
                          const float* __restrict__ bn_sum,
                          const float* __restrict__ bn_sumsq,
                          const float* __restrict__ gamma,
                          const float* __restrict__ beta,
                          const long long* __restrict__ batch,
                          unsigned* __restrict__ pooled_u, int N) {
  int i = blockIdx.x * blockDim.x + threadIdx.x;
  if (i >= N * CC) return;
  int c = i & (CC - 1);
  int n = i >> 7;
  float invN = 1.0f / (float)N;
  float mu = bn_sum[c] * invN;
  float var = bn_sumsq[c] * invN - mu * mu;
  float rs = rsqrtf(var + 1e-5f);
  float v = gamma[c] * (h_buf[i] - mu) * rs + beta[c];
  int g = (int)batch[n];
  atomicMax(&pooled_u[g * CC + c], f2u_mono(v));
}

__global__ void k_dec_pool(const unsigned* __restrict__ pooled_u,
                           float* __restrict__ out, int n) {
  int i = blockIdx.x * blockDim.x + threadIdx.x;
  if (i < n) out[i] = u2f_mono(pooled_u[i]);
}

// ---------------------------------------------------------------------------
extern "C" void kernel_launch(void* const* d_in, const int* in_sizes, int n_in,
                              void* d_out, int out_size, void* d_ws,
                              size_t ws_size, hipStream_t stream) {
  const float* x        = (const float*)d_in[0];
  const long long* ei   = (const long long*)d_in[1];
  const long long* batch= (const long long*)d_in[2];
  const float* Wg       = (const float*)d_in[3];
  const float* att_s    = (const float*)d_in[4];
  const float* att_d    = (const float*)d_in[5];
  const float* gat_bias = (const float*)d_in[6];
  const float* lin_w    = (const float*)d_in[7];
  const float* lin_b    = (const float*)d_in[8];
  const float* gamma    = (const float*)d_in[9];
  const float* beta     = (const float*)d_in[10];

  const int N = in_sizes[0] / DIN;
  const long long E = in_sizes[1] / 2;

  float* pooled_out = (float*)d_out;           // 64*128
  float* alpha_out  = pooled_out + 64 * CC;    // E*4

  // ---- workspace layout (floats) ----
  float* ws = (float*)d_ws;
  size_t off = 0;
  float* xh       = ws + off; off += (size_t)N * HC;
  float* lin_buf  = ws + off; off += (size_t)N * CC;
  float* a_s      = ws + off; off += (size_t)N * 4;
  float* a_d      = ws + off; off += (size_t)N * 4;
  float* e_buf    = ws + off; off += (size_t)E * 4;   // e, then alpha_un
  unsigned* emax_u= (unsigned*)(ws + off); off += (size_t)N * 4;
  float* denom    = ws + off; off += (size_t)N * 4;
  float* out_mean = ws + off; off += (size_t)N * CC;
  float* h_buf    = ws + off; off += (size_t)N * CC;
  float* bn_sum   = ws + off; off += CC;
  float* bn_sumsq = ws + off; off += CC;
  unsigned* pooled_u = (unsigned*)(ws + off); off += 64 * CC;

  hipMemsetAsync(emax_u, 0, (size_t)N * 4 * sizeof(unsigned), stream);
  hipMemsetAsync(denom, 0, (size_t)N * 4 * sizeof(float), stream);
  hipMemsetAsync(out_mean, 0, (size_t)N * CC * sizeof(float), stream);
  hipMemsetAsync(bn_sum, 0, CC * sizeof(float), stream);
  hipMemsetAsync(bn_sumsq, 0, CC * sizeof(float), stream);
  hipMemsetAsync(pooled_u, 0, 64 * CC * sizeof(unsigned), stream);

  // 1) WMMA projections: x_h = x @ W_gat ; lin_buf = x @ lin_w^T
  dim3 g1((N + 15) / 16, HC / 16);
  k_gemm8<<<g1, 32, 0, stream>>>(x, Wg, xh, N, HC, HC, 1);
  dim3 g2((N + 15) / 16, CC / 16);
  k_gemm8<<<g2, 32, 0, stream>>>(x, lin_w, lin_buf, N, CC, 1, DIN);

  // 2) attention logits
  int NH = N * 4;
  k_att<<<(NH + 255) / 256, 256, 0, stream>>>(xh, att_s, att_d, a_s, a_d, NH);

  // 3) edge logits + segment max
  int eb = (int)((E + 255) / 256);
  k_edge_max<<<eb, 256, 0, stream>>>(ei, a_s, a_d, e_buf, emax_u, E);

  // 4) exp + segment sum
  k_edge_exp<<<eb, 256, 0, stream>>>(ei, emax_u, e_buf, denom, E);

  // 5) normalize + weighted scatter (one wave per edge, 8 waves per block)
  int ab = (int)((E + 7) / 8);
  k_edge_aggr<<<ab, 256, 0, stream>>>(ei, xh, e_buf, denom, alpha_out,
                                      out_mean, E);

  // 6) residual + relu + BN partial sums
  int rows = 32;
  k_node_fin<<<(N + rows - 1) / rows, 128, 0, stream>>>(
      out_mean, lin_buf, gat_bias, lin_b, h_buf, bn_sum, bn_sumsq, N, rows);

  // 7) BN normalize + graph max-pool
  int tot = N * CC;
  k_bn_pool<<<(tot + 255) / 256, 256, 0, stream>>>(h_buf, bn_sum, bn_sumsq,
                                                   gamma, beta, batch,
                                                   pooled_u, N);

  // 8) decode pooled maxima into output
  k_dec_pool<<<(64 * CC + 255) / 256, 256, 0, stream>>>(pooled_u, pooled_out,
                                                        64 * CC);
}